// EntangledPairsEngine_12395275616275
// MI455X (gfx1250) — compile-verified
//
#include <hip/hip_runtime.h>
#include <hip/hip_bf16.h>
#include <math.h>

// ---------------------------------------------------------------------------
// EntangledPairsEngine for MI455X (gfx1250, wave32).
// Bandwidth-bound by bell_rotations (512 MB): streamed once, tanh on the fly,
// each element used for both rot@h_j and rot^T@h_i. GEMMs use exact fp32 WMMA
// (V_WMMA_F32_16X16X4_F32) with CDNA5 async global->LDS staging
// (GLOBAL_LOAD_ASYNC_TO_LDS_B128, ASYNCcnt) double-buffered so the next
// K-tile streams into LDS while WMMAs run on the current one.
// ---------------------------------------------------------------------------

typedef float v2f __attribute__((ext_vector_type(2)));
typedef float v8f __attribute__((ext_vector_type(8)));

#define N_CELLS 1024
#define IN_DIM  256
#define HID     512
#define OUT_DIM 256
#define INNER   128
#define N_PAIRS 512
#define KPAD    272   // 257 padded up to a multiple of 16

enum { MODE_BIAS = 0, MODE_RELU = 1, MODE_SUB = 2 };

__device__ __forceinline__ float sigf(float x) { return 1.f / (1.f + __expf(-x)); }

// CDNA5 async copy: per-lane 16B global -> LDS, tracked by ASYNCcnt.
// LDS operand is a VGPR holding the LDS byte address = addr[31:0] of the
// generic pointer (ISA 10.2).
__device__ __forceinline__ void async_g2l_b128(const float* gptr, float* lptr) {
    const uint32_t l = (uint32_t)(uintptr_t)lptr;
    asm volatile("global_load_async_to_lds_b128 %0, %1, off"
                 :: "v"(l), "v"(gptr) : "memory");
}
__device__ __forceinline__ void wait_asynccnt0() {
    asm volatile("s_wait_asynccnt 0x0" ::: "memory");
}

// ---------------------------------------------------------------------------
// Tiled fp32 WMMA GEMM:  C[m][n] = f( sum_k A[m][k] * W[n][k] + bias[n] )
// Block = 128 threads (4 waves), 64x64 C tile; each wave owns a 32x32 subtile
// as 2x2 v_wmma_f32_16x16x4_f32 accumulators. Double-buffered LDS staging via
// async global->LDS copies (no VGPR round-trip): stage s+1 streams in behind
// the 16 WMMAs of stage s; sync = s_wait_asynccnt 0 + workgroup barrier.
// LDS row stride 20 floats keeps 16B async stores / 8B fragment reads aligned.
// Requirements (guaranteed by caller): K%16==0, lda%4==0, ldw%4==0,
// grid = (N/64, M/64), M,N multiples of 64.
// ---------------------------------------------------------------------------
template <int K, int MODE>
__global__ __launch_bounds__(128)
void gemm_wmma_f32(const float* __restrict__ A, int lda,
                   const float* __restrict__ W, int ldw,
                   const float* __restrict__ bias,
                   float* __restrict__ C, int ldc,
                   const float* __restrict__ prev)
{
    static_assert(K % 16 == 0, "K must be a multiple of 16");
    constexpr int NSTAGE = K / 16;

    __shared__ __align__(16) float As[2][64 * 20];
    __shared__ __align__(16) float Ws[2][64 * 20];

    const int tid  = threadIdx.x;
    const int wave = tid >> 5;
    const int lane = tid & 31;
    const int half = lane >> 4;          // 0: lanes 0-15, 1: lanes 16-31
    const int lr   = lane & 15;
    const int wr   = (wave >> 1) * 32;   // wave row offset in 64x64 tile
    const int wc   = (wave & 1)  * 32;   // wave col offset
    const int m0 = blockIdx.y * 64;
    const int n0 = blockIdx.x * 64;

    // Staging map: thread copies rows (srow, srow+32) at 4-float column sc4.
    const int srow = tid >> 2;
    const int sc4  = (tid & 3) << 2;
    const float* Ab0 = A + (size_t)(m0 + srow) * lda + sc4;
    const float* Ab1 = Ab0 + (size_t)32 * lda;
    const float* Wb0 = W + (size_t)(n0 + srow) * ldw + sc4;
    const float* Wb1 = Wb0 + (size_t)32 * ldw;

    auto stage_async = [&](int k0, int buf) {
        async_g2l_b128(Ab0 + k0, &As[buf][srow * 20 + sc4]);
        async_g2l_b128(Ab1 + k0, &As[buf][(srow + 32) * 20 + sc4]);
        async_g2l_b128(Wb0 + k0, &Ws[buf][srow * 20 + sc4]);
        async_g2l_b128(Wb1 + k0, &Ws[buf][(srow + 32) * 20 + sc4]);
    };

    const v8f vzero = {0.f, 0.f, 0.f, 0.f, 0.f, 0.f, 0.f, 0.f};
    v8f acc[2][2];
    acc[0][0] = vzero; acc[0][1] = vzero; acc[1][0] = vzero; acc[1][1] = vzero;

    stage_async(0, 0);
    wait_asynccnt0();
    __syncthreads();

    for (int s = 0; s < NSTAGE; ++s) {
        const int buf = s & 1;
        // Next tile streams into the other buffer while WMMAs execute.
        if (s + 1 < NSTAGE) stage_async((s + 1) * 16, buf ^ 1);

        // ISA fragment layout for 16x16x4 f32: lanes 0-15 hold K={kb,kb+1},
        // lanes 16-31 hold K={kb+2,kb+3}; row/col index = lr.
#pragma unroll
        for (int kb = 0; kb < 16; kb += 4) {
            const int kk = kb + 2 * half;
            v2f a0 = *(const v2f*)&As[buf][(wr +      lr) * 20 + kk];
            v2f a1 = *(const v2f*)&As[buf][(wr + 16 + lr) * 20 + kk];
            v2f b0 = *(const v2f*)&Ws[buf][(wc +      lr) * 20 + kk];
            v2f b1 = *(const v2f*)&Ws[buf][(wc + 16 + lr) * 20 + kk];
            acc[0][0] = __builtin_amdgcn_wmma_f32_16x16x4_f32(false, a0, false, b0, (short)0, acc[0][0], false, false);
            acc[0][1] = __builtin_amdgcn_wmma_f32_16x16x4_f32(false, a0, false, b1, (short)0, acc[0][1], false, false);
            acc[1][0] = __builtin_amdgcn_wmma_f32_16x16x4_f32(false, a1, false, b0, (short)0, acc[1][0], false, false);
            acc[1][1] = __builtin_amdgcn_wmma_f32_16x16x4_f32(false, a1, false, b1, (short)0, acc[1][1], false, false);
        }

        if (s + 1 < NSTAGE) wait_asynccnt0();  // async writes landed in LDS
        __syncthreads();
    }

    // Epilogue. C/D layout: VGPR r -> M = tile + r + 8*half, N = tile + lr.
#pragma unroll
    for (int sm = 0; sm < 2; ++sm) {
#pragma unroll
        for (int sn = 0; sn < 2; ++sn) {
            const int nglob = n0 + wc + sn * 16 + lr;
            const float b = bias[nglob];
#pragma unroll
            for (int r = 0; r < 8; ++r) {
                const int mglob = m0 + wr + sm * 16 + r + 8 * half;
                float v = acc[sm][sn][r] + b;
                if constexpr (MODE == MODE_RELU) v = fmaxf(v, 0.f);
                const size_t off = (size_t)mglob * ldc + nglob;
                if constexpr (MODE == MODE_SUB) v = prev[off] - v;
                C[off] = v;
            }
        }
    }
}

// ---------------------------------------------------------------------------
// combined = [broadcast(x) | hiddens]  (1024 x 768)
// ---------------------------------------------------------------------------
__global__ __launch_bounds__(256)
void build_combined_kernel(const float* __restrict__ x, const float* __restrict__ h,
                           float* __restrict__ comb)
{
    const int idx = blockIdx.x * 256 + threadIdx.x;
    const int m = idx / 768, k = idx % 768;
    comb[idx] = (k < IN_DIM) ? x[k] : h[(size_t)m * HID + (k - IN_DIM)];
}

// tension[m] = mean_n output[m][n]^2
__global__ __launch_bounds__(256)
void tension_kernel(const float* __restrict__ outb, float* __restrict__ tension)
{
    __shared__ float red[256];
    const int m = blockIdx.x, t = threadIdx.x;
    const float v = outb[(size_t)m * OUT_DIM + t];
    red[t] = v * v;
    __syncthreads();
    for (int s = 128; s > 0; s >>= 1) { if (t < s) red[t] += red[t + s]; __syncthreads(); }
    if (t == 0) tension[m] = red[0] * (1.f / OUT_DIM);
}

// mem_input padded to 1024 x 272: [output | tension | zeros]
__global__ __launch_bounds__(256)
void build_mem_input_kernel(const float* __restrict__ output, const float* __restrict__ tension,
                            float* __restrict__ memin)
{
    const int idx = blockIdx.x * 256 + threadIdx.x;
    const int m = idx / KPAD, c = idx % KPAD;
    float v = 0.f;
    if (c < OUT_DIM)       v = output[(size_t)m * OUT_DIM + c];
    else if (c == OUT_DIM) v = tension[m];
    memin[idx] = v;
}

// w_ih (1536 x 257) -> zero-padded (1536 x 272)
__global__ __launch_bounds__(256)
void pad_wih_kernel(const float* __restrict__ w_ih, float* __restrict__ wpad)
{
    const int idx = blockIdx.x * 256 + threadIdx.x;
    const int n = idx / KPAD, c = idx % KPAD;
    wpad[idx] = (c < 257) ? w_ih[(size_t)n * 257 + c] : 0.f;
}

// GRU gate combine: new_h = (1-z)*tanh(i_n + r*h_n) + z*h
__global__ __launch_bounds__(256)
void gru_combine_kernel(const float* __restrict__ gi, const float* __restrict__ gh,
                        const float* __restrict__ h, float* __restrict__ newh)
{
    const int idx = blockIdx.x * 256 + threadIdx.x;
    const int m = idx >> 9;
    const int n = idx & 511;
    const size_t base = (size_t)m * (3 * HID);
    const float r  = sigf(gi[base + n]          + gh[base + n]);
    const float z  = sigf(gi[base + HID + n]    + gh[base + HID + n]);
    const float nn = tanhf(gi[base + 2*HID + n] + r * gh[base + 2*HID + n]);
    const float hv = h[(size_t)m * HID + n];
    newh[(size_t)m * HID + n] = (1.f - z) * nn + z * hv;
}

// ---------------------------------------------------------------------------
// Bell-pair kernel: one block per pair. Streams the 512x512 rot matrix ONCE
// (float4 coalesced + global_prefetch ahead), tanh on the fly, each element
// used for both mixed_i[a] += t*h_j[b] (lane-shuffle row reduction) and
// mixed_j[b] += t*h_i[a] (per-lane column accumulators).
// 16 waves = 4 row-groups x 4 col-strips; partials reduced through LDS.
// ---------------------------------------------------------------------------
__global__ __launch_bounds__(512)
void bell_kernel(const float* __restrict__ h, const float* __restrict__ ent,
                 const float* __restrict__ rot, float* __restrict__ hm)
{
    __shared__ float hi[HID], hj[HID];
    __shared__ float rowpart[HID * 4];   // [row][col-strip]
    __shared__ float colpart[4 * HID];   // [row-group][col]
    const int p = blockIdx.x;
    const int tid = threadIdx.x;
    hi[tid] = h[(size_t)(2 * p) * HID + tid];
    hj[tid] = h[(size_t)(2 * p + 1) * HID + tid];
    __syncthreads();

    const int wave = tid >> 5, lane = tid & 31;
    const int rg = wave >> 2;            // row group: 128 rows
    const int cs = wave & 3;             // col strip: 128 cols
    const int cbase = cs * 128 + lane * 4;
    const float4 hjv = *(const float4*)&hj[cbase];
    const float* rp = rot + (size_t)p * HID * HID;

    float aj0 = 0.f, aj1 = 0.f, aj2 = 0.f, aj3 = 0.f;
    const int a0 = rg * 128;
    for (int a = a0; a < a0 + 128; ++a) {
        // speculative prefetch 16 rows ahead (dropped on translation fail)
        __builtin_prefetch(&rp[(size_t)(a + 16) * HID + cbase], 0, 1);
        const float4 v = *(const float4*)&rp[(size_t)a * HID + cbase];
        const float t0 = tanhf(v.x), t1 = tanhf(v.y), t2 = tanhf(v.z), t3 = tanhf(v.w);
        const float hia = hi[a];
        aj0 += t0 * hia; aj1 += t1 * hia; aj2 += t2 * hia; aj3 += t3 * hia;
        float rsum = t0 * hjv.x + t1 * hjv.y + t2 * hjv.z + t3 * hjv.w;
#pragma unroll
        for (int msk = 16; msk >= 1; msk >>= 1) rsum += __shfl_xor(rsum, msk, 32);
        if (lane == 0) rowpart[a * 4 + cs] = rsum;
    }
    colpart[rg * HID + cbase + 0] = aj0;
    colpart[rg * HID + cbase + 1] = aj1;
    colpart[rg * HID + cbase + 2] = aj2;
    colpart[rg * HID + cbase + 3] = aj3;
    __syncthreads();

    const int c = tid;
    const float rs = rowpart[c * 4 + 0] + rowpart[c * 4 + 1] + rowpart[c * 4 + 2] + rowpart[c * 4 + 3];
    const float csum = colpart[c] + colpart[HID + c] + colpart[2 * HID + c] + colpart[3 * HID + c];
    const float e = sigf(ent[p]);
    const float oi = (1.f - e) * hi[c] + e * rs;
    const float oj = (1.f - e) * hj[c] + e * csum;
    hm[(size_t)(2 * p) * HID + c]     = oi;
    hm[(size_t)(2 * p + 1) * HID + c] = -oj;
}

// Faction sync: fmean + hout = 0.85*h + 0.15*fmean. grid (4 colchunks, 8 factions)
__global__ __launch_bounds__(128)
void faction_sync_kernel(const float* __restrict__ hm, float* __restrict__ hout,
                         float* __restrict__ fmean)
{
    const int f = blockIdx.y;
    const int c = blockIdx.x * 128 + threadIdx.x;
    float s = 0.f;
    for (int j = 0; j < 128; ++j) s += hm[(size_t)(f * 128 + j) * HID + c];
    const float fm = s * (1.f / 128.f);
    fmean[f * HID + c] = fm;
    for (int j = 0; j < 128; ++j) {
        const size_t off = (size_t)(f * 128 + j) * HID + c;
        hout[off] = 0.85f * hm[off] + 0.15f * fm;
    }
}

// Debate (step > 5): global_opinion = mean_f fmean; rows [0,32) of each faction.
__global__ __launch_bounds__(128)
void faction_debate_kernel(float* __restrict__ hout, const float* __restrict__ fmean,
                           const int* __restrict__ step)
{
    if (*step <= 5) return;
    const int f = blockIdx.y;
    const int c = blockIdx.x * 128 + threadIdx.x;
    float go = 0.f;
    for (int ff = 0; ff < 8; ++ff) go += fmean[ff * HID + c];
    go *= (1.f / 8.f);
    for (int j = 0; j < 32; ++j) {
        const size_t off = (size_t)(f * 128 + j) * HID + c;
        hout[off] = 0.85f * hout[off] + 0.15f * go;
    }
}

// Softmax(tension) -> weighted sum of output -> head matvec + avg_tension.
__global__ __launch_bounds__(256)
void tail_kernel(const float* __restrict__ tension, const float* __restrict__ output,
                 const float* __restrict__ head_w, const float* __restrict__ head_b,
                 float* __restrict__ out)
{
    __shared__ float tw[N_CELLS];
    __shared__ float red[256];
    __shared__ float co[OUT_DIM];
    const int t = threadIdx.x;
    float lmax = -1e30f, lsum = 0.f;
    for (int i = t; i < N_CELLS; i += 256) {
        const float v = tension[i];
        tw[i] = v;
        lmax = fmaxf(lmax, v);
        lsum += v;
    }
    red[t] = lmax;
    __syncthreads();
    for (int s = 128; s > 0; s >>= 1) { if (t < s) red[t] = fmaxf(red[t], red[t + s]); __syncthreads(); }
    const float mx = red[0];
    __syncthreads();
    red[t] = lsum;
    __syncthreads();
    for (int s = 128; s > 0; s >>= 1) { if (t < s) red[t] += red[t + s]; __syncthreads(); }
    if (t == 0) out[OUT_DIM] = red[0] * (1.f / N_CELLS);   // avg_tension
    __syncthreads();
    float lexp = 0.f;
    for (int i = t; i < N_CELLS; i += 256) {
        const float e = expf(tw[i] - mx);
        tw[i] = e;
        lexp += e;
    }
    red[t] = lexp;
    __syncthreads();
    for (int s = 128; s > 0; s >>= 1) { if (t < s) red[t] += red[t + s]; __syncthreads(); }
    const float inv = 1.f / red[0];
    __syncthreads();
    float acc = 0.f;
    for (int m = 0; m < N_CELLS; ++m) acc += tw[m] * output[(size_t)m * OUT_DIM + t];
    co[t] = acc * inv;
    __syncthreads();
    float pv = head_b[t];
    for (int n = 0; n < OUT_DIM; ++n) pv += head_w[(size_t)t * OUT_DIM + n] * co[n];
    out[t] = pv;   // pred
}

// ---------------------------------------------------------------------------
extern "C" void kernel_launch(void* const* d_in, const int* in_sizes, int n_in,
                              void* d_out, int out_size, void* d_ws, size_t ws_size,
                              hipStream_t stream)
{
    const float* x       = (const float*)d_in[0];
    const float* hiddens = (const float*)d_in[1];
    const float* W1a     = (const float*)d_in[2];
    const float* b1a     = (const float*)d_in[3];
    const float* W2a     = (const float*)d_in[4];
    const float* b2a     = (const float*)d_in[5];
    const float* W1g     = (const float*)d_in[6];
    const float* b1g     = (const float*)d_in[7];
    const float* W2g     = (const float*)d_in[8];
    const float* b2g     = (const float*)d_in[9];
    const float* w_ih    = (const float*)d_in[10];
    const float* w_hh    = (const float*)d_in[11];
    const float* b_ih    = (const float*)d_in[12];
    const float* b_hh    = (const float*)d_in[13];
    const float* head_w  = (const float*)d_in[14];
    const float* head_b  = (const float*)d_in[15];
    const float* ent     = (const float*)d_in[16];
    const float* bell    = (const float*)d_in[17];
    const int*   step    = (const int*)d_in[18];

    float* ws = (float*)d_ws;
    float* combined = ws;                        // 1024*768   = 786432
    float* relu1a   = combined + 786432;         // 1024*128   = 131072
    float* relu1g   = relu1a   + 131072;         // 131072
    float* outbuf   = relu1g   + 131072;         // 1024*256   = 262144
    float* tension  = outbuf   + 262144;         // 1024
    float* memin    = tension  + 1024;           // 1024*272   = 278528 (padded)
    float* wihpad   = memin    + 278528;         // 1536*272   = 417792 (padded)
    float* gi       = wihpad   + 417792;         // 1024*1536  = 1572864
    float* gh       = gi       + 1572864;        // 1572864
    float* newh     = gh       + 1572864;        // 1024*512   = 524288
    float* hmix     = newh     + 524288;         // 524288
    float* fmean    = hmix     + 524288;         // 8*512      = 4096
    // total ~6.21M floats (~24.8 MB)

    float* out  = (float*)d_out;                 // [pred 256 | avg_tension 1 | h 1024*512]
    float* hout = out + 257;

    // 1. combined = [x | hiddens]; pad w_ih (independent, can go early)
    build_combined_kernel<<<(N_CELLS * 768) / 256, 256, 0, stream>>>(x, hiddens, combined);
    pad_wih_kernel<<<(1536 * KPAD) / 256, 256, 0, stream>>>(w_ih, wihpad);
    // 2-3. MLP layer 1 (relu) for a and g
    gemm_wmma_f32<768, MODE_RELU><<<dim3(INNER / 64, N_CELLS / 64), 128, 0, stream>>>(
        combined, 768, W1a, 768, b1a, relu1a, INNER, nullptr);
    gemm_wmma_f32<768, MODE_RELU><<<dim3(INNER / 64, N_CELLS / 64), 128, 0, stream>>>(
        combined, 768, W1g, 768, b1g, relu1g, INNER, nullptr);
    // 4-5. MLP layer 2; second pass fuses output = a - g
    gemm_wmma_f32<INNER, MODE_BIAS><<<dim3(OUT_DIM / 64, N_CELLS / 64), 128, 0, stream>>>(
        relu1a, INNER, W2a, INNER, b2a, outbuf, OUT_DIM, nullptr);
    gemm_wmma_f32<INNER, MODE_SUB><<<dim3(OUT_DIM / 64, N_CELLS / 64), 128, 0, stream>>>(
        relu1g, INNER, W2g, INNER, b2g, outbuf, OUT_DIM, outbuf);
    // 6. tension
    tension_kernel<<<N_CELLS, 256, 0, stream>>>(outbuf, tension);
    // 7. mem_input (padded to K=272)
    build_mem_input_kernel<<<(N_CELLS * KPAD) / 256, 256, 0, stream>>>(outbuf, tension, memin);
    // 8-9. GRU GEMMs (both branch-free)
    gemm_wmma_f32<KPAD, MODE_BIAS><<<dim3((3 * HID) / 64, N_CELLS / 64), 128, 0, stream>>>(
        memin, KPAD, wihpad, KPAD, b_ih, gi, 3 * HID, nullptr);
    gemm_wmma_f32<HID, MODE_BIAS><<<dim3((3 * HID) / 64, N_CELLS / 64), 128, 0, stream>>>(
        hiddens, HID, w_hh, HID, b_hh, gh, 3 * HID, nullptr);
    // 10. GRU gates
    gru_combine_kernel<<<(N_CELLS * HID) / 256, 256, 0, stream>>>(gi, gh, hiddens, newh);
    // 11. Bell pairs (bandwidth-dominant: streams 512 MB once)
    bell_kernel<<<N_PAIRS, 512, 0, stream>>>(newh, ent, bell, hmix);
    // 12-13. Faction sync + debate into d_out h region
    faction_sync_kernel<<<dim3(4, 8), 128, 0, stream>>>(hmix, hout, fmean);
    faction_debate_kernel<<<dim3(4, 8), 128, 0, stream>>>(hout, fmean, step);
    // 14. softmax-weighted readout + head
    tail_kernel<<<1, 256, 0, stream>>>(tension, outbuf, head_w, head_b, out);
}